// NTMWrapper_68324339745372
// MI455X (gfx1250) — compile-verified
//
#include <hip/hip_runtime.h>
#include <cmath>

// ---------------------------------------------------------------------------
// NTM content-addressed write+read, MI455X (gfx1250, wave32).
// Bandwidth-bound: M0 is 256MB; single-pass online softmax => ~11us roofline.
// Head projections use V_WMMA_F32_16X16X4_F32 (exact f32 GEMM on tensor path).
// ---------------------------------------------------------------------------

typedef float v2f __attribute__((ext_vector_type(2)));
typedef float v8f __attribute__((ext_vector_type(8)));

#define B_SZ    64
#define D_INV   256
#define N_ROWS  16384
#define W_LEN   64
#define EPSF    1e-16f
#define SPLITS  16
#define WAVES   8
#define REC_STRIDE 132   // {M,Z1,Z2,pad} + R1[64] + R2[64]

// float offsets into workspace
#define WS_K    0
#define WS_E    4096
#define WS_A    8192
#define WS_BETA 12288
#define WS_PART 12352

// ---------------------------------------------------------------------------
// Kernel 1: head GEMMs  H = x @ W (+bias, activation).
// Blocks 0..47: one wave per 16x16 tile of {k, erase, add}; K=256 via 64 WMMAs.
// Block 48: beta head (N=1) + softplus.
// ---------------------------------------------------------------------------
__global__ __launch_bounds__(32) void ntm_heads(
    const float* __restrict__ x,
    const float* __restrict__ Wk, const float* __restrict__ bk,
    const float* __restrict__ Wb, const float* __restrict__ bb,
    const float* __restrict__ We, const float* __restrict__ be,
    const float* __restrict__ Wa, const float* __restrict__ ba,
    float* __restrict__ ws)
{
    const int blk  = blockIdx.x;
    const int lane = threadIdx.x;

    if (blk == 48) {                      // beta = softplus(x @ W_beta + b_beta)
        for (int rep = 0; rep < 2; ++rep) {
            const int b = lane + 32 * rep;
            float acc = bb[0];
            for (int kk = 0; kk < D_INV; ++kk)
                acc = fmaf(x[b * D_INV + kk], Wb[kk], acc);
            ws[WS_BETA + b] = fmaxf(acc, 0.0f) + log1pf(__expf(-fabsf(acc)));
        }
        return;
    }

    const int mat  = blk >> 4;            // 0:k(tanh) 1:erase(sigmoid) 2:add(tanh)
    const int tile = blk & 15;
    const int rt   = (tile >> 2) << 4;    // row-tile base (batch dim)
    const int ct   = (tile & 3) << 4;     // col-tile base (W_LEN dim)
    const float* W    = (mat == 0) ? Wk : (mat == 1) ? We : Wa;
    const float* bias = (mat == 0) ? bk : (mat == 1) ? be : ba;
    float* out = ws + ((mat == 0) ? WS_K : (mat == 1) ? WS_E : WS_A);

    const int l  = lane & 15;
    const int hi = lane >> 4;

    v8f c = {};
    for (int k0 = 0; k0 < D_INV; k0 += 4) {
        const int ka = k0 + 2 * hi;       // lanes 0-15: K={k0,k0+1}; 16-31: K={k0+2,k0+3}
        v2f a, bm;
        a.x  = x[(rt + l) * D_INV + ka];
        a.y  = x[(rt + l) * D_INV + ka + 1];
        bm.x = W[ka       * W_LEN + ct + l];
        bm.y = W[(ka + 1) * W_LEN + ct + l];
        c = __builtin_amdgcn_wmma_f32_16x16x4_f32(false, a, false, bm,
                                                  (short)0, c, false, false);
    }
#pragma unroll
    for (int v = 0; v < 8; ++v) {
        const int row = rt + v + 8 * hi;  // C layout: VGPR v = M {v, v+8}
        const int col = ct + l;
        float val = c[v] + bias[col];
        val = (mat == 1) ? (1.0f / (1.0f + __expf(-val))) : tanhf(val);
        out[row * W_LEN + col] = val;
    }
}

// ---------------------------------------------------------------------------
// Kernel 2: single streaming pass over M0 with online softmax.
// Half-wave (16 lanes x float4) = one 256B row (fully coalesced b128 loads).
// Accumulates M (running max of beta*sim), Z1=sum e, Z2=sum e^2,
// R1=sum e*row, R2=sum e^2*row; partials to workspace.
// ---------------------------------------------------------------------------
__global__ __launch_bounds__(256) void ntm_scan(
    const float* __restrict__ M0,
    const float* __restrict__ ws,
    float* __restrict__ part)
{
    const int b     = blockIdx.x / SPLITS;
    const int split = blockIdx.x % SPLITS;
    const int tid   = threadIdx.x;
    const int wave  = tid >> 5;
    const int lane  = tid & 31;
    const int l     = lane & 15;
    const int hi    = lane >> 4;

    // key: load, L2-normalize on the fly (cheap, avoids an extra kernel)
    const float4 kv = *(const float4*)(ws + WS_K + b * W_LEN + 4 * l);
    float kq = kv.x * kv.x + kv.y * kv.y + kv.z * kv.z + kv.w * kv.w;
    kq += __shfl_xor(kq, 1, 32);
    kq += __shfl_xor(kq, 2, 32);
    kq += __shfl_xor(kq, 4, 32);
    kq += __shfl_xor(kq, 8, 32);
    const float kinv = 1.0f / (sqrtf(kq) + EPSF);
    const float k4x = kv.x * kinv, k4y = kv.y * kinv;
    const float k4z = kv.z * kinv, k4w = kv.w * kinv;
    const float beta = ws[WS_BETA + b];

    const int rowsPerWG   = N_ROWS / SPLITS;        // 1024
    const int rowsPerWave = rowsPerWG / WAVES;      // 128 (2 rows/iter, 64 iters)
    const int rowBase     = split * rowsPerWG + wave * rowsPerWave;
    const float* base = M0 + ((size_t)b * N_ROWS + rowBase + hi) * (size_t)W_LEN + 4 * l;

    float R1x = 0, R1y = 0, R1z = 0, R1w = 0;
    float R2x = 0, R2y = 0, R2z = 0, R2w = 0;
    float Z1 = 0, Z2 = 0, Mrun = -__builtin_inff();

    const int ITER = rowsPerWave / 2;               // 64

#define LD(ii) (*(const float4*)(base + (size_t)(ii) * 128))

#define PROC(v)                                                            \
    {                                                                      \
        float d_ = k4x * v.x + k4y * v.y + k4z * v.z + k4w * v.w;          \
        float q_ = v.x * v.x + v.y * v.y + v.z * v.z + v.w * v.w;          \
        d_ += __shfl_xor(d_, 1, 32);  q_ += __shfl_xor(q_, 1, 32);         \
        d_ += __shfl_xor(d_, 2, 32);  q_ += __shfl_xor(q_, 2, 32);         \
        d_ += __shfl_xor(d_, 4, 32);  q_ += __shfl_xor(q_, 4, 32);         \
        d_ += __shfl_xor(d_, 8, 32);  q_ += __shfl_xor(q_, 8, 32);         \
        float s_  = beta * d_ / (sqrtf(q_) + EPSF);                        \
        float nm_ = fmaxf(Mrun, s_);                                       \
        float c1_ = __expf(Mrun - nm_);                                    \
        float e_  = __expf(s_ - nm_);                                      \
        Mrun = nm_;                                                        \
        float c2_ = c1_ * c1_, e2_ = e_ * e_;                              \
        Z1 = fmaf(Z1, c1_, e_);  Z2 = fmaf(Z2, c2_, e2_);                  \
        R1x = fmaf(R1x, c1_, e_ * v.x);  R1y = fmaf(R1y, c1_, e_ * v.y);   \
        R1z = fmaf(R1z, c1_, e_ * v.z);  R1w = fmaf(R1w, c1_, e_ * v.w);   \
        R2x = fmaf(R2x, c2_, e2_ * v.x); R2y = fmaf(R2y, c2_, e2_ * v.y);  \
        R2z = fmaf(R2z, c2_, e2_ * v.z); R2w = fmaf(R2w, c2_, e2_ * v.w);  \
    }

    // 4-deep software pipeline: keep 4 b128 loads in flight per lane.
    float4 v0 = LD(0), v1 = LD(1), v2 = LD(2), v3 = LD(3);
    for (int g = 0; g < ITER / 4; ++g) {
        const int nx = (g + 1) * 4;
        const int i0 = (nx + 0 < ITER) ? nx + 0 : 0;
        const int i1 = (nx + 1 < ITER) ? nx + 1 : 0;
        const int i2 = (nx + 2 < ITER) ? nx + 2 : 0;
        const int i3 = (nx + 3 < ITER) ? nx + 3 : 0;
        float4 n0 = LD(i0), n1 = LD(i1), n2 = LD(i2), n3 = LD(i3);
        PROC(v0); PROC(v1); PROC(v2); PROC(v3);
        v0 = n0; v1 = n1; v2 = n2; v3 = n3;
    }
#undef PROC
#undef LD

    // merge the two half-wave accumulators (partner lane = lane ^ 16)
    {
        const float Mo = __shfl_xor(Mrun, 16, 32);
        const float nm = fmaxf(Mrun, Mo);
        const float c1 = __expf(Mrun - nm);
        const float c2 = c1 * c1;
        Mrun = nm;
        Z1 *= c1;  Z2 *= c2;
        R1x *= c1; R1y *= c1; R1z *= c1; R1w *= c1;
        R2x *= c2; R2y *= c2; R2z *= c2; R2w *= c2;
        Z1  += __shfl_xor(Z1, 16, 32);
        Z2  += __shfl_xor(Z2, 16, 32);
        R1x += __shfl_xor(R1x, 16, 32); R1y += __shfl_xor(R1y, 16, 32);
        R1z += __shfl_xor(R1z, 16, 32); R1w += __shfl_xor(R1w, 16, 32);
        R2x += __shfl_xor(R2x, 16, 32); R2y += __shfl_xor(R2y, 16, 32);
        R2z += __shfl_xor(R2z, 16, 32); R2w += __shfl_xor(R2w, 16, 32);
    }

    float* rec = part + (size_t)((b * SPLITS + split) * WAVES + wave) * REC_STRIDE;
    if (lane == 0) { rec[0] = Mrun; rec[1] = Z1; rec[2] = Z2; rec[3] = 0.0f; }
    if (lane < 16) {
        float4 r1 = make_float4(R1x, R1y, R1z, R1w);
        float4 r2 = make_float4(R2x, R2y, R2z, R2w);
        *(float4*)(rec + 4  + 4 * l) = r1;
        *(float4*)(rec + 68 + 4 * l) = r2;
    }
}

// ---------------------------------------------------------------------------
// Kernel 3: combine 128 partials per batch, emit
// read[b,m] = R1/Z1 - erase*R2/Z1^2 + add*Z2/Z1^2.
// ---------------------------------------------------------------------------
__global__ __launch_bounds__(64) void ntm_combine(
    const float* __restrict__ part,
    const float* __restrict__ ws,
    float* __restrict__ out)
{
    const int b = blockIdx.x;
    const int t = threadIdx.x;
    const float* P = part + (size_t)b * (SPLITS * WAVES) * REC_STRIDE;

    float Mg = -__builtin_inff();
    for (int p = 0; p < SPLITS * WAVES; ++p)
        Mg = fmaxf(Mg, P[(size_t)p * REC_STRIDE]);

    float Z1 = 0, Z2 = 0, R1 = 0, R2 = 0;
    for (int p = 0; p < SPLITS * WAVES; ++p) {
        const float* rec = P + (size_t)p * REC_STRIDE;
        const float c  = __expf(rec[0] - Mg);
        const float c2 = c * c;
        Z1 = fmaf(rec[1],      c,  Z1);
        Z2 = fmaf(rec[2],      c2, Z2);
        R1 = fmaf(rec[4 + t],  c,  R1);
        R2 = fmaf(rec[68 + t], c2, R2);
    }
    const float inv  = 1.0f / Z1;
    const float inv2 = inv * inv;
    const float er = ws[WS_E + b * W_LEN + t];
    const float ad = ws[WS_A + b * W_LEN + t];
    out[b * W_LEN + t] = R1 * inv - er * (R2 * inv2) + ad * (Z2 * inv2);
}

// ---------------------------------------------------------------------------
extern "C" void kernel_launch(void* const* d_in, const int* in_sizes, int n_in,
                              void* d_out, int out_size, void* d_ws, size_t ws_size,
                              hipStream_t stream)
{
    const float* x  = (const float*)d_in[0];
    const float* Wk = (const float*)d_in[1];
    const float* bk = (const float*)d_in[2];
    const float* Wb = (const float*)d_in[3];
    const float* bb = (const float*)d_in[4];
    const float* We = (const float*)d_in[5];
    const float* be = (const float*)d_in[6];
    const float* Wa = (const float*)d_in[7];
    const float* ba = (const float*)d_in[8];
    const float* M0 = (const float*)d_in[9];
    float* out  = (float*)d_out;
    float* ws   = (float*)d_ws;
    float* part = ws + WS_PART;

    hipLaunchKernelGGL(ntm_heads,   dim3(49),             dim3(32),  0, stream,
                       x, Wk, bk, Wb, bb, We, be, Wa, ba, ws);
    hipLaunchKernelGGL(ntm_scan,    dim3(B_SZ * SPLITS),  dim3(256), 0, stream,
                       M0, ws, part);
    hipLaunchKernelGGL(ntm_combine, dim3(B_SZ),           dim3(64),  0, stream,
                       part, ws, out);
}